// DynamicGraphUpdater_54374285967980
// MI455X (gfx1250) — compile-verified
//
#include <hip/hip_runtime.h>

// Problem constants (match reference shapes)
#define R_ 8
#define N_ 100000
#define H_ 128
#define T_ 64

typedef __attribute__((ext_vector_type(2))) float v2f;
typedef __attribute__((ext_vector_type(8))) float v8f;

// ---------------- workspace init ----------------
__global__ void init_ws_kernel(int* __restrict__ last_pos, int* __restrict__ edge_idx,
                               int* __restrict__ counts, int* __restrict__ cursor,
                               int n_lp, int n_ei) {
    int i = blockIdx.x * blockDim.x + threadIdx.x;
    if (i < n_lp) last_pos[i] = -1;
    if (i < n_ei) edge_idx[i] = -1;
    if (i < R_) { counts[i] = 0; cursor[i] = 0; }
}

// ---------------- last-edge-wins + per-relation counts ----------------
__global__ void scatter_count_kernel(const int* __restrict__ rel, const int* __restrict__ src,
                                     int* __restrict__ last_pos, int* __restrict__ counts, int E) {
    int e = blockIdx.x * blockDim.x + threadIdx.x;
    if (e >= E) return;
    int r = rel[e];
    atomicMax(&last_pos[r * N_ + src[e]], e);
    atomicAdd(&counts[r], 1);
}

// ---------------- padded prefix offsets (tiny) ----------------
__global__ void offsets_kernel(const int* __restrict__ counts, int* __restrict__ offsets,
                               int* __restrict__ cursor) {
    if (threadIdx.x == 0 && blockIdx.x == 0) {
        int off = 0;
        for (int r = 0; r < R_; ++r) {
            offsets[r] = off;
            cursor[r]  = off;
            off += ((counts[r] + 15) >> 4) << 4;   // pad region to multiple of 16
        }
        offsets[R_] = off;
    }
}

// ---------------- bin edges by relation ----------------
__global__ void bin_kernel(const int* __restrict__ rel, int* __restrict__ cursor,
                           int* __restrict__ edge_idx, int E) {
    int e = blockIdx.x * blockDim.x + threadIdx.x;
    if (e >= E) return;
    int pos = atomicAdd(&cursor[rel[e]], 1);
    edge_idx[pos] = e;
}

// ---------------- bulk copy prev -> out (bandwidth floor) ----------------
__global__ void copy_kernel(const float4* __restrict__ in, float4* __restrict__ out, int n4) {
    int i = blockIdx.x * blockDim.x + threadIdx.x;
    if (i < n4) out[i] = in[i];
}

// ---------------- GRU via f32 WMMA: one block = one 16-edge tile ----------------
__launch_bounds__(256)
__global__ void gru_wmma_kernel(
    const float* __restrict__ prev,     // [R,N,H]
    const float* __restrict__ encW,     // [2R,H,T]
    const float* __restrict__ encB,     // [2R,H]
    const float* __restrict__ Wih,      // [R,3H,2H]
    const float* __restrict__ Whh,      // [R,3H,H]
    const float* __restrict__ bih,      // [R,3H]
    const float* __restrict__ bhh,      // [R,3H]
    const int*   __restrict__ src,
    const int*   __restrict__ dst,
    const int*   __restrict__ rel,
    const int*   __restrict__ t_edge,
    const int*   __restrict__ cur_time, // scalar
    const int*   __restrict__ edge_idx,
    const int*   __restrict__ offsets,
    const int*   __restrict__ last_pos,
    float*       __restrict__ out)      // [R,N,H]
{
    __shared__ float xA[16][260];       // [edge][ x(0:128)=enc+u_j | x(128:256)=u_i ], padded stride
    __shared__ int sh_e[16], sh_src[16], sh_dst[16], sh_bkt[16], sh_win[16];
    __shared__ int sh_rel;

    const int tid = threadIdx.x;
    const int tileBase = blockIdx.x * 16;
    const int total = offsets[R_];
    if (tileBase >= total) return;      // uniform per block

    // ---- phase 0: tile metadata ----
    if (tid == 0) {
        int e0 = edge_idx[tileBase];    // slot 0 of every live tile is a valid edge
        sh_rel = (e0 >= 0) ? rel[e0] : 0;
    }
    if (tid < 16) {
        int e = edge_idx[tileBase + tid];
        sh_e[tid] = e;
        if (e >= 0) {
            sh_src[tid] = src[e];
            sh_dst[tid] = dst[e];
            int b = cur_time[0] - t_edge[e];
            sh_bkt[tid] = (b < T_ - 1) ? b : (T_ - 1);
        } else {
            sh_src[tid] = 0; sh_dst[tid] = 0; sh_bkt[tid] = 0;
        }
    }
    __syncthreads();
    const int r = sh_rel;
    if (tid < 16) {
        int e = sh_e[tid];
        sh_win[tid] = (e >= 0 && last_pos[r * N_ + sh_src[tid]] == e) ? 1 : 0;
    }

    // ---- phase 1: build x tile in LDS ----
    {
        int m  = tid >> 4;
        int hb = (tid & 15) * 8;
        int e  = sh_e[m];
        if (e >= 0) {
            const float* pj = prev + (long)(r * N_ + sh_dst[m]) * H_;
            const float* pi = prev + (long)(r * N_ + sh_src[m]) * H_;
            const float* ew = encW + (long)r * H_ * T_;
            const float* eb = encB + (long)r * H_;
            int bk = sh_bkt[m];
            #pragma unroll
            for (int q = 0; q < 8; ++q) {
                int h = hb + q;
                float enc = ew[h * T_ + bk] + eb[h];
                xA[m][h]        = enc + pj[h];
                xA[m][H_ + h]   = pi[h];
            }
        } else {
            #pragma unroll
            for (int q = 0; q < 8; ++q) {
                int h = hb + q;
                xA[m][h] = 0.0f; xA[m][H_ + h] = 0.0f;
            }
        }
    }
    __syncthreads();

    // ---- phase 2: WMMA GEMMs; wave w owns hidden columns [16w,16w+16) of all 3 gates ----
    const int wave = tid >> 5;          // 0..7 (wave32)
    const int lane = tid & 31;
    const int half = lane >> 4;         // 0: K+{0,1}, 1: K+{2,3}
    const int lr   = lane & 15;         // A: row M; B/D: column N
    const int colBase = wave * 16;
    const int ksh  = 2 * half;

    const float* WihR = Wih + (long)r * (3 * H_) * (2 * H_);
    const float* WhhR = Whh + (long)r * (3 * H_) * H_;
    const float* rowIr = WihR + (long)(0 * H_ + colBase + lr) * (2 * H_);
    const float* rowIz = WihR + (long)(1 * H_ + colBase + lr) * (2 * H_);
    const float* rowIn = WihR + (long)(2 * H_ + colBase + lr) * (2 * H_);
    const float* rowHr = WhhR + (long)(0 * H_ + colBase + lr) * H_;
    const float* rowHz = WhhR + (long)(1 * H_ + colBase + lr) * H_;
    const float* rowHn = WhhR + (long)(2 * H_ + colBase + lr) * H_;

    v8f aIr = {0.f,0.f,0.f,0.f,0.f,0.f,0.f,0.f};
    v8f aIz = aIr, aIn = aIr, aHr = aIr, aHz = aIr, aHn = aIr;

    // gi = x @ W_ih^T : K = 256
    for (int k = 0; k < 2 * H_; k += 4) {
        v2f a  = *(const v2f*)(&xA[lr][k + ksh]);
        v2f br = *(const v2f*)(rowIr + k + ksh);
        v2f bz = *(const v2f*)(rowIz + k + ksh);
        v2f bn = *(const v2f*)(rowIn + k + ksh);
        aIr = __builtin_amdgcn_wmma_f32_16x16x4_f32(false, a, false, br, (short)0, aIr, false, false);
        aIz = __builtin_amdgcn_wmma_f32_16x16x4_f32(false, a, false, bz, (short)0, aIz, false, false);
        aIn = __builtin_amdgcn_wmma_f32_16x16x4_f32(false, a, false, bn, (short)0, aIn, false, false);
    }
    // gh = h_prev @ W_hh^T : K = 128, h_prev lives at xA[:][128..255]
    for (int k = 0; k < H_; k += 4) {
        v2f a  = *(const v2f*)(&xA[lr][H_ + k + ksh]);
        v2f br = *(const v2f*)(rowHr + k + ksh);
        v2f bz = *(const v2f*)(rowHz + k + ksh);
        v2f bn = *(const v2f*)(rowHn + k + ksh);
        aHr = __builtin_amdgcn_wmma_f32_16x16x4_f32(false, a, false, br, (short)0, aHr, false, false);
        aHz = __builtin_amdgcn_wmma_f32_16x16x4_f32(false, a, false, bz, (short)0, aHz, false, false);
        aHn = __builtin_amdgcn_wmma_f32_16x16x4_f32(false, a, false, bn, (short)0, aHn, false, false);
    }

    // biases (same for all 8 M-rows held by this lane)
    const int ob = colBase + lr;
    float bIr = bih[r * 3 * H_ + 0 * H_ + ob];
    float bIz = bih[r * 3 * H_ + 1 * H_ + ob];
    float bIn = bih[r * 3 * H_ + 2 * H_ + ob];
    float bHr = bhh[r * 3 * H_ + 0 * H_ + ob];
    float bHz = bhh[r * 3 * H_ + 1 * H_ + ob];
    float bHn = bhh[r * 3 * H_ + 2 * H_ + ob];

    // ---- phase 3: GRU nonlinearity + winner scatter ----
    #pragma unroll
    for (int v = 0; v < 8; ++v) {
        int mm = v + 8 * half;          // edge slot (D layout: VGPR v -> M=v or v+8)
        float ir = aIr[v] + bIr, hr = aHr[v] + bHr;
        float iz = aIz[v] + bIz, hz = aHz[v] + bHz;
        float in_ = aIn[v] + bIn, hn = aHn[v] + bHn;
        float rg = 1.0f / (1.0f + __expf(-(ir + hr)));
        float zg = 1.0f / (1.0f + __expf(-(iz + hz)));
        float ng = tanhf(in_ + rg * hn);
        float hp = xA[mm][H_ + ob];
        float cand = (1.0f - zg) * ng + zg * hp;
        if (sh_win[mm]) {
            out[(long)(r * N_ + sh_src[mm]) * H_ + ob] = cand;
        }
    }
}

extern "C" void kernel_launch(void* const* d_in, const int* in_sizes, int n_in,
                              void* d_out, int out_size, void* d_ws, size_t ws_size,
                              hipStream_t stream) {
    const float* prev   = (const float*)d_in[0];
    const float* encW   = (const float*)d_in[1];
    const float* encB   = (const float*)d_in[2];
    const float* Wih    = (const float*)d_in[3];
    const float* Whh    = (const float*)d_in[4];
    const float* bih    = (const float*)d_in[5];
    const float* bhh    = (const float*)d_in[6];
    const int*   src    = (const int*)d_in[7];
    const int*   dst    = (const int*)d_in[8];
    const int*   rel    = (const int*)d_in[9];
    const int*   t_edge = (const int*)d_in[10];
    const int*   ctime  = (const int*)d_in[11];
    float*       outp   = (float*)d_out;

    const int E = in_sizes[7];
    const int n_lp = R_ * N_;               // 800000
    const int n_ei = E + 16 * R_;           // padded bin array

    int* last_pos = (int*)d_ws;
    int* edge_idx = last_pos + n_lp;
    int* counts   = edge_idx + n_ei;
    int* offsets  = counts + R_;
    int* cursor   = offsets + (R_ + 1);

    // 1. init workspace
    init_ws_kernel<<<(n_lp + 255) / 256, 256, 0, stream>>>(last_pos, edge_idx, counts, cursor, n_lp, n_ei);
    // 2. last-edge-wins + counts
    scatter_count_kernel<<<(E + 255) / 256, 256, 0, stream>>>(rel, src, last_pos, counts, E);
    // 3. padded offsets
    offsets_kernel<<<1, 32, 0, stream>>>(counts, offsets, cursor);
    // 4. bin by relation
    bin_kernel<<<(E + 255) / 256, 256, 0, stream>>>(rel, cursor, edge_idx, E);
    // 5. bulk copy prev -> out
    const int n4 = (R_ * N_ * H_) / 4;
    copy_kernel<<<(n4 + 255) / 256, 256, 0, stream>>>((const float4*)prev, (float4*)outp, n4);
    // 6. GRU tiles (upper bound on padded tile count; blocks past offsets[R] exit)
    const int maxTiles = (E + 15) / 16 + R_;
    gru_wmma_kernel<<<maxTiles, 256, 0, stream>>>(
        prev, encW, encB, Wih, Whh, bih, bhh,
        src, dst, rel, t_edge, ctime,
        edge_idx, offsets, last_pos, outp);
}